// SkipGram_8315056685705
// MI455X (gfx1250) — compile-verified
//
#include <hip/hip_runtime.h>
#include <hip/hip_bf16.h>
#include <math.h>

typedef __attribute__((ext_vector_type(2))) float v2f;
typedef __attribute__((ext_vector_type(4))) float v4f;
typedef __attribute__((ext_vector_type(8))) float v8f;

// log_sigmoid(x) = min(x,0) - log1p(exp(-|x|)).
// Scores are bounded (|x| <= 8 for this problem: sums of <=8 products of
// uniforms), so exp never under/overflows and the hardware transcendentals
// (v_exp_f32 / v_log_f32) are accurate to ~1e-4 relative here.
__device__ __forceinline__ float log_sigmoid_fast(float x) {
    return fminf(x, 0.0f) - __logf(1.0f + __expf(-fabsf(x)));
}

// One wave (32 lanes) handles 16 samples.
// p_m[k] = E[f, f+O_k] * E[c, f+O_k] * [(f+O_k - c) in OFFSETS],  k = 0..7
// score_m = sum_k p_m[k], computed as D = P(16x8) x Ones(8x16) via two
// chained V_WMMA_F32_16X16X4_F32; every column of D equals the row-sum, so
// lane 0 holds scores 0..7 in its 8 acc VGPRs and lane 16 holds scores 8..15.
__global__ void __launch_bounds__(256) skipgram_wmma_kernel(
    const int* __restrict__ focus,
    const int* __restrict__ context,
    const float* __restrict__ E,
    float* __restrict__ out,
    int B)
{
    constexpr int LD = 8228;  // EMBED_SIZE row stride
    const int OFF[8] = {0, 12, 19, 24, 28, 31, 34, 36};
    const unsigned long long OMASK =
        (1ull << 0) | (1ull << 12) | (1ull << 19) | (1ull << 24) |
        (1ull << 28) | (1ull << 31) | (1ull << 34) | (1ull << 36);

    const int tid      = threadIdx.x;
    const int lane     = tid & 31;
    const int wave     = tid >> 5;
    const int waveBase = blockIdx.x * 128 + wave * 16;  // 16 samples / wave
    const int m        = lane & 15;                     // sample within wave
    const int hi       = lane >> 4;                     // 0: K{0,1,4,5}  1: K{2,3,6,7}

    int s = waveBase + m;
    if (s >= B) s = B - 1;  // clamp loads; EXEC must stay all-1s for WMMA

    const int f = focus[s];
    const int c = context[s];
    const size_t frow = (size_t)f * LD;
    const size_t crow = (size_t)c * LD;

    // This lane's 4 of the 8 K-slots (A-matrix f32 16x4 layout:
    //   VGPR0 = K{0 | 2}, VGPR1 = K{1 | 3} split by lane half).
    // Both loads are always in-bounds (col <= 8227 < LD), so the offset-set
    // membership test is a branchless {0,1} multiply — no EXEC churn.
    float p[4];
#pragma unroll
    for (int q = 0; q < 4; ++q) {
        const int k   = (q & 1) + 2 * hi + (q >> 1) * 4;
        const int col = f + OFF[k];
        const unsigned t = (unsigned)(col - c);  // valid iff an offset value
        const float vmask =
            (t < 64u) ? (float)((OMASK >> t) & 1ull) : 0.0f;
        const float af = E[frow + col];
        const float bc = E[crow + col];
        p[q] = af * bc * vmask;
    }

    const v2f A_lo  = {p[0], p[1]};   // K = 0..3 slice
    const v2f A_hi  = {p[2], p[3]};   // K = 4..7 slice
    const v2f Bones = {1.0f, 1.0f};   // all-ones B: reduce-K + broadcast

    v8f acc = {};
    // 8-arg form: (neg_a, A, neg_b, B, c_mod, C, reuse_a, reuse_b)
    acc = __builtin_amdgcn_wmma_f32_16x16x4_f32(
        false, A_lo, false, Bones, (short)0, acc, false, false);
    acc = __builtin_amdgcn_wmma_f32_16x16x4_f32(
        false, A_hi, false, Bones, (short)0, acc, false, false);

    // Apply log_sigmoid uniformly (uniform per half-wave; cheap with hw
    // transcendentals), then only lanes 0 and 16 store their 8 results as
    // two 128-bit stores (base is a multiple of 8 -> 32B aligned).
    v8f r;
#pragma unroll
    for (int v = 0; v < 8; ++v) r[v] = log_sigmoid_fast(acc[v]);

    if ((lane & 15) == 0) {
        const int base = waveBase + hi * 8;
        if (base + 7 < B) {
            *(v4f*)(out + base)     = v4f{r[0], r[1], r[2], r[3]};
            *(v4f*)(out + base + 4) = v4f{r[4], r[5], r[6], r[7]};
        } else {
#pragma unroll
            for (int v = 0; v < 8; ++v) {
                const int o = base + v;
                if (o < B) out[o] = r[v];
            }
        }
    }
}

extern "C" void kernel_launch(void* const* d_in, const int* in_sizes, int n_in,
                              void* d_out, int out_size, void* d_ws, size_t ws_size,
                              hipStream_t stream) {
    (void)n_in; (void)out_size; (void)d_ws; (void)ws_size;
    const int*   focus   = (const int*)d_in[0];
    const int*   context = (const int*)d_in[1];
    const float* E       = (const float*)d_in[2];
    float*       out     = (float*)d_out;

    const int B = in_sizes[0];                 // 8192 samples
    const int blocks = (B + 127) / 128;        // 128 samples per 256-thread block
    skipgram_wmma_kernel<<<blocks, 256, 0, stream>>>(focus, context, E, out, B);
}